// GCNLayer_39049842655816
// MI455X (gfx1250) — compile-verified
//
#include <hip/hip_runtime.h>

#define N_NODES 100000
#define N_EDGES 1600000
#define D_IN    128
#define D_OUT   64
#define M_TILES ((N_NODES + 15) / 16)   // 6250, exact: 6250*16 = 100000

typedef __attribute__((ext_vector_type(2))) float v2f;
typedef __attribute__((ext_vector_type(8))) float v8f;

// ---------------------------------------------------------------------------
// Kernel 1: h = x @ W^T  via V_WMMA_F32_16X16X4_F32 (wave32).
// One wave computes a 16(M) x 64(N) tile: A fragment loaded once, reused for
// 4 N-tiles (4 accumulators). K=128 swept in steps of 4.
//
// A 16x4 f32 layout: lanes 0-15 -> M=lane, v0=K0 v1=K1; lanes 16-31 -> K2,K3.
// B 4x16 f32 layout (mirrored): lanes 0-15 -> N=lane, v0=B[0][n] v1=B[1][n];
//                               lanes 16-31 -> B[2][n], B[3][n].
// Since B = W^T, B[k][n] = W[n][k] -> contiguous float2 along K from W rows.
// C/D 16x16 f32: vgpr i holds M=i (lanes 0-15) / M=i+8 (lanes 16-31), N=lane%16.
// ---------------------------------------------------------------------------
__global__ __launch_bounds__(256) void gcn_gemm(const float* __restrict__ x,
                                                const float* __restrict__ W,
                                                float* __restrict__ h) {
    const int lane   = threadIdx.x & 31;
    const int wave   = threadIdx.x >> 5;
    const int m_tile = blockIdx.x * 8 + wave;
    if (m_tile >= M_TILES) return;          // wave-uniform: EXEC stays all-1s

    const int m0 = m_tile * 16;
    const int r  = lane & 15;               // A: M row half / B: N column
    const int kh = (lane >> 4) * 2;         // K sub-offset: 0 or 2

    const float* __restrict__ xp  = x + (size_t)(m0 + r) * D_IN + kh;
    const float* __restrict__ wp0 = W + (size_t)( 0 + r) * D_IN + kh;
    const float* __restrict__ wp1 = W + (size_t)(16 + r) * D_IN + kh;
    const float* __restrict__ wp2 = W + (size_t)(32 + r) * D_IN + kh;
    const float* __restrict__ wp3 = W + (size_t)(48 + r) * D_IN + kh;

    v8f acc0 = {}, acc1 = {}, acc2 = {}, acc3 = {};

#pragma unroll
    for (int k = 0; k < D_IN; k += 4) {
        v2f a  = *(const v2f*)(xp  + k);
        v2f b0 = *(const v2f*)(wp0 + k);
        v2f b1 = *(const v2f*)(wp1 + k);
        v2f b2 = *(const v2f*)(wp2 + k);
        v2f b3 = *(const v2f*)(wp3 + k);
        acc0 = __builtin_amdgcn_wmma_f32_16x16x4_f32(false, a, false, b0, (short)0, acc0, false, false);
        acc1 = __builtin_amdgcn_wmma_f32_16x16x4_f32(false, a, false, b1, (short)0, acc1, false, false);
        acc2 = __builtin_amdgcn_wmma_f32_16x16x4_f32(false, a, false, b2, (short)0, acc2, false, false);
        acc3 = __builtin_amdgcn_wmma_f32_16x16x4_f32(false, a, false, b3, (short)0, acc3, false, false);
    }

    const int mbase = m0 + ((lane >> 4) << 3);
    const int col   = lane & 15;
#pragma unroll
    for (int i = 0; i < 8; ++i) {
        float* hp = h + (size_t)(mbase + i) * D_OUT + col;
        hp[0]  = acc0[i];
        hp[16] = acc1[i];
        hp[32] = acc2[i];
        hp[48] = acc3[i];
    }
}

// ---------------------------------------------------------------------------
// Kernel 2a: deg[i] = 1.0  (self-loop contribution to degree)
// ---------------------------------------------------------------------------
__global__ void deg_init(float* __restrict__ deg) {
    int i = blockIdx.x * blockDim.x + threadIdx.x;
    if (i < N_NODES) deg[i] = 1.0f;
}

// Kernel 2b: deg[src[e]] += 1 for each edge (hardware f32 atomic, no return)
__global__ void deg_accum(const long long* __restrict__ src,
                          float* __restrict__ deg) {
    int e = blockIdx.x * blockDim.x + threadIdx.x;
    if (e < N_EDGES) {
        int s = (int)src[e];
        unsafeAtomicAdd(&deg[s], 1.0f);
    }
}

// Kernel 2c: dis[i] = rsqrt(deg[i])  (deg >= 1 always, in-place)
__global__ void deg_rsqrt(float* __restrict__ deg) {
    int i = blockIdx.x * blockDim.x + threadIdx.x;
    if (i < N_NODES) deg[i] = __frsqrt_rn(deg[i]);
}

// ---------------------------------------------------------------------------
// Kernel 3: out = h * dis^2  (self-loop message; also initializes d_out)
// ---------------------------------------------------------------------------
__global__ void self_loop(const float* __restrict__ h,
                          const float* __restrict__ dis,
                          float* __restrict__ out) {
    int t = blockIdx.x * blockDim.x + threadIdx.x;
    if (t < N_NODES * D_OUT) {
        int i = t >> 6;
        float w = dis[i];
        out[t] = h[t] * (w * w);
    }
}

// ---------------------------------------------------------------------------
// Kernel 4: out[dst] += h[src] * dis[src]*dis[dst].  64 threads per edge,
// one channel per thread: coalesced 256B gather + 256B of f32 atomics.
// ---------------------------------------------------------------------------
__global__ __launch_bounds__(256) void aggregate(const float* __restrict__ h,
                                                 const float* __restrict__ dis,
                                                 const long long* __restrict__ src,
                                                 const long long* __restrict__ dst,
                                                 float* __restrict__ out) {
    int t = blockIdx.x * 256 + threadIdx.x;
    int e = t >> 6;
    int c = t & 63;
    if (e < N_EDGES) {
        int s = (int)src[e];
        int d = (int)dst[e];
        float w = dis[s] * dis[d];
        unsafeAtomicAdd(&out[(size_t)d * D_OUT + c],
                        h[(size_t)s * D_OUT + c] * w);
    }
}

// ---------------------------------------------------------------------------
// Kernel 5: ReLU in place
// ---------------------------------------------------------------------------
__global__ void relu_k(float* __restrict__ out) {
    int t = blockIdx.x * blockDim.x + threadIdx.x;
    if (t < N_NODES * D_OUT) out[t] = fmaxf(out[t], 0.0f);
}

// ---------------------------------------------------------------------------
extern "C" void kernel_launch(void* const* d_in, const int* in_sizes, int n_in,
                              void* d_out, int out_size, void* d_ws, size_t ws_size,
                              hipStream_t stream) {
    const float*     x  = (const float*)d_in[0];
    const long long* ei = (const long long*)d_in[1];   // int64 edge_index [2, E]
    const float*     W  = (const float*)d_in[2];
    float*           out = (float*)d_out;

    // workspace: h [N_NODES*D_OUT floats] then deg/dis [N_NODES floats]
    float* h   = (float*)d_ws;
    float* dis = (float*)((char*)d_ws + (size_t)N_NODES * D_OUT * sizeof(float));

    const long long* src = ei;
    const long long* dst = ei + N_EDGES;

    // 1) h = x @ W^T  (WMMA)
    gcn_gemm<<<(M_TILES + 7) / 8, 256, 0, stream>>>(x, W, h);

    // 2) degree -> deg_inv_sqrt
    deg_init <<<(N_NODES + 255) / 256, 256, 0, stream>>>(dis);
    deg_accum<<<(N_EDGES + 255) / 256, 256, 0, stream>>>(src, dis);
    deg_rsqrt<<<(N_NODES + 255) / 256, 256, 0, stream>>>(dis);

    // 3) out = self-loop messages (initializes d_out)
    self_loop<<<(N_NODES * D_OUT + 255) / 256, 256, 0, stream>>>(h, dis, out);

    // 4) scatter-add edge messages
    aggregate<<<((size_t)N_EDGES * D_OUT + 255) / 256, 256, 0, stream>>>(h, dis, src, dst, out);

    // 5) ReLU
    relu_k<<<(N_NODES * D_OUT + 255) / 256, 256, 0, stream>>>(out);
}